// RetinaNetPostProcessor_55791625175774
// MI455X (gfx1250) — compile-verified
//
#include <hip/hip_runtime.h>
#include <stdint.h>

// ---------------- problem constants (from reference) ----------------
#define NIMG   2
#define AA     9          // anchors per cell
#define CCC    80         // classes
#define HHH    100
#define WWW    152
#define PIX    (HHH*WWW)              // 15200
#define LLL    (PIX*AA)               // 136800 anchors per image
#define PER_IMG (AA*CCC*PIX)          // 10,944,000 cls elements / image
#define TOTAL_CLS (NIMG*PER_IMG)      // 21,888,000
#define TOPK   1000
#define POSTK  100
#define CAP    4096                   // candidate buffer per image
#define NBINS  2048
#define CHUNK  8192                   // floats per TDM tile (32 KB)
#define TLOGIT (-2.94443897917f)      // log(0.05/0.95): sigmoid(x)>0.05 <=> x>TLOGIT
#define BINW   (12.0f/(float)NBINS)
#define CLIPV  4.13516655674f         // log(1000/16)
#define IMGW   1216.0f
#define IMGH   800.0f
#define OFFMUL 1218.0f                // max(W,H)+2
#define NMS_T  0.5f
#define MINSZ  0.0f

typedef uint32_t u32;
typedef uint64_t u64;
typedef __attribute__((ext_vector_type(4))) u32 u32x4;
typedef __attribute__((ext_vector_type(8))) u32 u32x8;

// ---------------- kernel 1: zero the atomically-updated ws regions ----------------
__global__ void k_init(u32* __restrict__ hist, u32* __restrict__ cnt) {
  int t = blockIdx.x * blockDim.x + threadIdx.x;
  if (t < NIMG * NBINS) hist[t] = 0u;
  if (t < NIMG)         cnt[t]  = 0u;
}

// ---- TDM issue helper: 1-D tile of n f32 elements, global 'ga' -> LDS 'ldsAddr' ----
__device__ __forceinline__ void tdm_load_1d(u64 ga, u32 ldsAddr, u32 n) {
  // D# group 0: count=1 (valid), lds_addr, 57-bit global addr, type=2 ("image")
  u32x4 g0 = { 1u,
               ldsAddr,
               (u32)ga,
               (u32)((ga >> 32) & 0x1FFFFFFu) | (2u << 30) };
  // D# group 1: wg_mask=0, data_size=2 (4B); tensor_dim0=n, tensor_dim1=1,
  // tile_dim0=n (<=65535), tile_dim1/2 unused, tensor_dim0_stride=n
  u32x8 g1 = { (2u << 16),
               (n & 0xFFFFu) << 16,
               ((n >> 16) & 0xFFFFu) | (1u << 16),
               (n & 0xFFFFu) << 16,
               0u,
               n,
               0u, 0u };
  u32x4 z2 = { 0u, 0u, 0u, 0u };
  u32x4 z3 = { 0u, 0u, 0u, 0u };
  asm volatile("tensor_load_to_lds %0, %1, %2, %3"
               :: "s"(g0), "s"(g1), "s"(z2), "s"(z3) : "memory");
}

// ---------------- kernel 2: streaming histogram, double-buffered TDM through LDS ----------------
// Pipeline (wave 0 drives the Tensor Data Mover, tracked by TENSORcnt):
//   issue tile0 -> [ wait tile k | issue tile k+1 into alternate buffer | consume tile k ] ...
// so HBM DMA of tile k+1 overlaps LDS histogramming of tile k.
__global__ void k_hist(const float* __restrict__ cls, u32* __restrict__ ghist) {
  __shared__ float tile[2][CHUNK];      // 2 x 32 KB
  __shared__ u32   lh[NIMG * NBINS];    // 16 KB local histogram
  for (int i = threadIdx.x; i < NIMG * NBINS; i += blockDim.x) lh[i] = 0u;
  __syncthreads();

  const u64 base     = (u64)(uintptr_t)cls;
  const u32 ldsAddr0 = (u32)(uintptr_t)&tile[0][0];  // low 32 bits of flat addr == LDS offset
  const u32 ldsAddr1 = (u32)(uintptr_t)&tile[1][0];
  const int nChunks  = (TOTAL_CLS + CHUNK - 1) / CHUNK;
  const int stride   = gridDim.x;

  int ch = blockIdx.x;
  if (ch >= nChunks) {
    // still must participate in the final flush (no barriers skipped: none remain)
    for (int i = threadIdx.x; i < NIMG * NBINS; i += blockDim.x) {
      u32 v = lh[i]; if (v) atomicAdd(&ghist[i], v);
    }
    return;
  }

  // prologue: DMA first tile into buffer 0
  if (threadIdx.x < 32) {
    u32 n0 = ((u32)TOTAL_CLS - (u32)ch * CHUNK) < (u32)CHUNK
           ? ((u32)TOTAL_CLS - (u32)ch * CHUNK) : (u32)CHUNK;
    tdm_load_1d(base + (u64)((u32)ch * CHUNK) * 4ull, ldsAddr0, n0);
  }

  int p = 0;
  for (; ch < nChunks; ch += stride, p ^= 1) {
    const u32 start = (u32)ch * (u32)CHUNK;
    const u32 n = ((u32)TOTAL_CLS - start) < (u32)CHUNK ? ((u32)TOTAL_CLS - start) : (u32)CHUNK;

    if (threadIdx.x < 32) __builtin_amdgcn_s_wait_tensorcnt(0);  // tile[p] ready
    __syncthreads();   // all waves: tile[p] visible; tile[p^1] consumers (prev iter) done

    // kick DMA for the next chunk into the alternate buffer while we consume tile[p]
    int nch = ch + stride;
    if (threadIdx.x < 32 && nch < nChunks) {
      u32 ns = (u32)nch * (u32)CHUNK;
      u32 nn = ((u32)TOTAL_CLS - ns) < (u32)CHUNK ? ((u32)TOTAL_CLS - ns) : (u32)CHUNK;
      tdm_load_1d(base + (u64)ns * 4ull, p ? ldsAddr0 : ldsAddr1, nn);
    }

    for (u32 i = threadIdx.x; i < n; i += blockDim.x) {
      float x = tile[p][i];
      if (x > TLOGIT) {
        u32 gi  = start + i;
        int img = (gi >= (u32)PER_IMG) ? 1 : 0;
        int b = (int)((x - TLOGIT) * (1.0f / BINW));
        if (b > NBINS - 1) b = NBINS - 1;
        atomicAdd(&lh[img * NBINS + b], 1u);
      }
    }
    __syncthreads();   // tile[p] fully consumed before it is reused two iterations later
  }

  for (int i = threadIdx.x; i < NIMG * NBINS; i += blockDim.x) {
    u32 v = lh[i];
    if (v) atomicAdd(&ghist[i], v);
  }
}

// ---------------- kernel 3: pick cutoff bin so that >= TOPK elements survive ----------------
__global__ void k_cutoff(const u32* __restrict__ hist, float* __restrict__ cutoff) {
  int img = threadIdx.x;
  if (img >= NIMG) return;
  u32 cum = 0; int cb = 0;
  for (int b = NBINS - 1; b >= 0; --b) {
    cum += hist[img * NBINS + b];
    if (cum >= (u32)TOPK) { cb = b; break; }
  }
  cutoff[img] = TLOGIT + (float)cb * BINW;  // bin lower edge; collects exactly bins >= cb
}

// ---------------- kernel 4: collect candidates above cutoff (2nd streaming pass) ----------------
__global__ void k_collect(const float* __restrict__ cls, const float* __restrict__ cutoff,
                          u32* __restrict__ cnt, float* __restrict__ csc, int* __restrict__ cidx) {
  const float c0 = cutoff[0], c1 = cutoff[1];
  const int nvec = TOTAL_CLS / 4;
  for (int v = blockIdx.x * blockDim.x + threadIdx.x; v < nvec; v += gridDim.x * blockDim.x) {
    float4 x4 = ((const float4*)cls)[v];
    __builtin_prefetch(cls + (size_t)v * 4 + 16384, 0, 0);  // global_prefetch_b8
#pragma unroll
    for (int k = 0; k < 4; ++k) {
      float x = (&x4.x)[k];
      int o = v * 4 + k;
      int img = (o >= PER_IMG) ? 1 : 0;
      float cv = img ? c1 : c0;
      if (x >= cv && x > TLOGIT) {
        int oo = img ? (o - PER_IMG) : o;
        int w  = oo % WWW; int r = oo / WWW;
        int h  = r % HHH;  r /= HHH;           // r = a*C + c
        int a  = r / CCC;  int c = r % CCC;
        int flat = ((h * WWW + w) * AA + a) * CCC + c;   // reference (L*C) flat index
        u32 p = atomicAdd(&cnt[img], 1u);
        if (p < (u32)CAP) {
          csc [img * CAP + p] = 1.0f / (1.0f + expf(-x));
          cidx[img * CAP + p] = flat;
        }
      }
    }
  }
}

// ---------------- in-LDS bitonic sort, descending by key, tie -> smaller val first ----------------
template <int NN>
__device__ inline void bitonic_desc(float* key, int* val) {
  for (int k = 2; k <= NN; k <<= 1) {
    for (int j = k >> 1; j > 0; j >>= 1) {
      for (int i = threadIdx.x; i < NN; i += blockDim.x) {
        int ixj = i ^ j;
        if (ixj > i) {
          float ki = key[i], kj = key[ixj];
          int   vi = val[i], vj = val[ixj];
          bool iBefore = (ki > kj) || ((ki == kj) && (vi < vj));
          bool desc = ((i & k) == 0);
          if (desc != iBefore) { key[i] = kj; key[ixj] = ki; val[i] = vj; val[ixj] = vi; }
        }
      }
      __syncthreads();
    }
  }
}

// ---------------- kernel 5: sort candidates, emit exact top-1000 (matches top_k tie rules) ----------
__global__ void k_sortcand(const u32* __restrict__ cnt, const float* __restrict__ csc,
                           const int* __restrict__ cidx, float* __restrict__ tv, int* __restrict__ ti) {
  const int img = blockIdx.x;
  __shared__ float key[CAP];   // 16 KB
  __shared__ int   val[CAP];   // 16 KB  (320 KB WGP LDS makes this cheap)
  u32 n = cnt[img]; if (n > (u32)CAP) n = CAP;
  for (int i = threadIdx.x; i < CAP; i += blockDim.x) {
    if ((u32)i < n) { key[i] = csc[img * CAP + i]; val[i] = cidx[img * CAP + i]; }
    else            { key[i] = -1e30f;             val[i] = i; }
  }
  __syncthreads();
  bitonic_desc<CAP>(key, val);
  for (int i = threadIdx.x; i < TOPK; i += blockDim.x) {
    float s = key[i]; int id = val[i];
    if (s <= -1e29f) { s = -1.0f; id = 0; }   // fewer than 1000 above threshold -> pad like masked -1
    tv[img * TOPK + i] = s; ti[img * TOPK + i] = id;
  }
}

// ---------------- kernel 6: decode boxes, min-size mask, stable re-sort by (sc desc, pos asc) -------
__global__ void k_decode(const float* __restrict__ tv, const int* __restrict__ ti,
                         const float* __restrict__ reg, const float* __restrict__ anc,
                         float* __restrict__ bpre, int* __restrict__ clspre,
                         float* __restrict__ bs, int* __restrict__ clss, float* __restrict__ scs) {
  const int img = blockIdx.x;
  __shared__ float sk[1024];
  __shared__ int   sp[1024];
  for (int i = threadIdx.x; i < 1024; i += blockDim.x) {
    float sc = -2.0f;                           // sentinel below any real score / -1
    if (i < TOPK) {
      float v  = tv[img * TOPK + i];
      int   fl = ti[img * TOPK + i];
      int loc = fl / CCC;
      int c   = fl % CCC + 1;
      int a   = loc % AA;
      int pix = loc / AA;
      int ph  = pix / WWW, pw = pix % WWW;
      float rel[4];
#pragma unroll
      for (int j = 0; j < 4; ++j)
        rel[j] = reg[((((size_t)img * AA + a) * 4 + j) * HHH + ph) * WWW + pw];
      const float* an = anc + (size_t)loc * 4;
      float aw = an[2] - an[0] + 1.f, ah = an[3] - an[1] + 1.f;
      float cx = an[0] + 0.5f * aw,   cy = an[1] + 0.5f * ah;
      float dx = rel[0] / 10.0f, dy = rel[1] / 10.0f;
      float dw = fminf(rel[2] / 5.0f, CLIPV), dh = fminf(rel[3] / 5.0f, CLIPV);
      float pcx = dx * aw + cx, pcy = dy * ah + cy;
      float pww = expf(dw) * aw, phh = expf(dh) * ah;
      float x1 = pcx - 0.5f * pww,       y1 = pcy - 0.5f * phh;
      float x2 = pcx + 0.5f * pww - 1.f, y2 = pcy + 0.5f * phh - 1.f;
      x1 = fminf(fmaxf(x1, 0.f), IMGW - 1.f); x2 = fminf(fmaxf(x2, 0.f), IMGW - 1.f);
      y1 = fminf(fmaxf(y1, 0.f), IMGH - 1.f); y2 = fminf(fmaxf(y2, 0.f), IMGH - 1.f);
      bool ks = (x2 - x1 + 1.f >= MINSZ) && (y2 - y1 + 1.f >= MINSZ);
      sc = ks ? v : -1.0f;
      float* bp = bpre + ((size_t)img * TOPK + i) * 4;
      bp[0] = x1; bp[1] = y1; bp[2] = x2; bp[3] = y2;
      clspre[img * TOPK + i] = c;
    }
    sk[i] = sc; sp[i] = i;
  }
  __syncthreads();
  bitonic_desc<1024>(sk, sp);        // stable wrt original pos for equal scores
  for (int r = threadIdx.x; r < TOPK; r += blockDim.x) {
    int p = sp[r];
    const float* bp = bpre + ((size_t)img * TOPK + p) * 4;
    float*       bd = bs   + ((size_t)img * TOPK + r) * 4;
    bd[0] = bp[0]; bd[1] = bp[1]; bd[2] = bp[2]; bd[3] = bp[3];
    clss[img * TOPK + r] = clspre[img * TOPK + p];
    scs [img * TOPK + r] = sk[r];
  }
}

// ---------------- kernel 7: 1000x1000 IoU suppression bitmask (32 u32 words / row) -----------------
__global__ void k_iou(const float* __restrict__ bs, const int* __restrict__ clss,
                      u32* __restrict__ mask) {
  const int img = blockIdx.y;
  int t = blockIdx.x * blockDim.x + threadIdx.x;
  if (t >= TOPK * 32) return;
  int i  = t >> 5;         // row
  int jw = t & 31;         // column word
  const float* bi = bs + ((size_t)img * TOPK + i) * 4;
  float offi = (float)clss[img * TOPK + i] * OFFMUL;
  float ix1 = bi[0] + offi, iy1 = bi[1] + offi, ix2 = bi[2] + offi, iy2 = bi[3] + offi;
  float ai = (ix2 - ix1 + 1.f) * (iy2 - iy1 + 1.f);
  u32 bits = 0u;
  int j0 = jw * 32;
  for (int b = 0; b < 32; ++b) {
    int j = j0 + b;
    if (j > i && j < TOPK) {
      const float* bj = bs + ((size_t)img * TOPK + j) * 4;
      float offj = (float)clss[img * TOPK + j] * OFFMUL;
      float jx1 = bj[0] + offj, jy1 = bj[1] + offj, jx2 = bj[2] + offj, jy2 = bj[3] + offj;
      float aj = (jx2 - jx1 + 1.f) * (jy2 - jy1 + 1.f);
      float ltx = fmaxf(ix1, jx1), lty = fmaxf(iy1, jy1);
      float rbx = fminf(ix2, jx2), rby = fminf(iy2, jy2);
      float iw = fmaxf(rbx - ltx + 1.f, 0.f), ih = fmaxf(rby - lty + 1.f, 0.f);
      float inter = iw * ih;
      float iou = inter / (ai + aj - inter);
      if (iou > NMS_T) bits |= (1u << b);
    }
  }
  mask[((size_t)img * TOPK + i) * 32 + jw] = bits;
}

// ---------------- kernel 8: sequential greedy NMS scan, one wave32 per image -----------------------
__global__ void k_scan(const float* __restrict__ scs, const u32* __restrict__ mask,
                       u32* __restrict__ keep) {
  const int img = blockIdx.x;
  __shared__ u32 removed[32];
  __shared__ int kflag;
  removed[threadIdx.x] = 0u;
  __syncthreads();
  for (int i = 0; i < TOPK; ++i) {
    if (threadIdx.x == 0)
      kflag = (scs[img * TOPK + i] >= 0.0f) && !((removed[i >> 5] >> (i & 31)) & 1u);
    __syncthreads();
    if (kflag) removed[threadIdx.x] |= mask[((size_t)img * TOPK + i) * 32 + threadIdx.x];
    if (threadIdx.x == 0) keep[img * TOPK + i] = kflag ? 1u : 0u;
    __syncthreads();
  }
}

// ---------------- kernel 9: emit top-100 (kept in order, then -1 fillers by ascending pos) ---------
__global__ void k_final(const float* __restrict__ bs, const int* __restrict__ clss,
                        const float* __restrict__ scs, const u32* __restrict__ keep,
                        float* __restrict__ out) {
  const int img = blockIdx.x;
  int slot = 0;
  for (int pass = 0; pass < 2 && slot < POSTK; ++pass) {
    for (int i = 0; i < TOPK && slot < POSTK; ++i) {
      bool kp = keep[img * TOPK + i] != 0u;
      if ((pass == 0) == kp) {
        const float* b = bs + ((size_t)img * TOPK + i) * 4;
        float* ob = out + ((size_t)img * POSTK + slot) * 4;
        ob[0] = b[0]; ob[1] = b[1]; ob[2] = b[2]; ob[3] = b[3];
        out[NIMG * POSTK * 4 + img * POSTK + slot] = kp ? scs[img * TOPK + i] : -1.0f;
        out[NIMG * POSTK * 5 + img * POSTK + slot] = (float)clss[img * TOPK + i];
        slot++;
      }
    }
  }
}

// -------------------------------- launcher --------------------------------
extern "C" void kernel_launch(void* const* d_in, const int* in_sizes, int n_in,
                              void* d_out, int out_size, void* d_ws, size_t ws_size,
                              hipStream_t stream) {
  (void)in_sizes; (void)n_in; (void)out_size; (void)ws_size;
  const float* cls = (const float*)d_in[0];   // (2, 720, 100, 152)
  const float* reg = (const float*)d_in[1];   // (2,  36, 100, 152)
  const float* anc = (const float*)d_in[2];   // (136800, 4)
  float* out = (float*)d_out;                 // 1200 floats: boxes|scores|labels

  char* w = (char*)d_ws;
  auto take = [&](size_t bytes) { char* p = w; w += (bytes + 255) & ~(size_t)255; return p; };
  u32*   hist   = (u32*)  take((size_t)NIMG * NBINS * sizeof(u32));
  float* cutoff = (float*)take((size_t)NIMG * sizeof(float));
  u32*   cnt    = (u32*)  take((size_t)NIMG * sizeof(u32));
  float* csc    = (float*)take((size_t)NIMG * CAP * sizeof(float));
  int*   cidx   = (int*)  take((size_t)NIMG * CAP * sizeof(int));
  float* tv     = (float*)take((size_t)NIMG * TOPK * sizeof(float));
  int*   ti     = (int*)  take((size_t)NIMG * TOPK * sizeof(int));
  float* bpre   = (float*)take((size_t)NIMG * TOPK * 4 * sizeof(float));
  int*   clspre = (int*)  take((size_t)NIMG * TOPK * sizeof(int));
  float* bs     = (float*)take((size_t)NIMG * TOPK * 4 * sizeof(float));
  int*   clss   = (int*)  take((size_t)NIMG * TOPK * sizeof(int));
  float* scs    = (float*)take((size_t)NIMG * TOPK * sizeof(float));
  u32*   mask   = (u32*)  take((size_t)NIMG * TOPK * 32 * sizeof(u32));
  u32*   keep   = (u32*)  take((size_t)NIMG * TOPK * sizeof(u32));
  // total ws use ~450 KB

  k_init    <<<(NIMG * NBINS + 255) / 256, 256, 0, stream>>>(hist, cnt);
  k_hist    <<<668, 256, 0, stream>>>(cls, hist);     // 2672 chunks, 4 per block, 2-deep TDM pipeline
  k_cutoff  <<<1, NIMG, 0, stream>>>(hist, cutoff);
  k_collect <<<2048, 256, 0, stream>>>(cls, cutoff, cnt, csc, cidx);
  k_sortcand<<<NIMG, 256, 0, stream>>>(cnt, csc, cidx, tv, ti);
  k_decode  <<<NIMG, 256, 0, stream>>>(tv, ti, reg, anc, bpre, clspre, bs, clss, scs);
  dim3 giou((TOPK * 32 + 255) / 256, NIMG);
  k_iou     <<<giou, 256, 0, stream>>>(bs, clss, mask);
  k_scan    <<<NIMG, 32, 0, stream>>>(scs, mask, keep);
  k_final   <<<NIMG, 1, 0, stream>>>(bs, clss, scs, keep, out);
}